// PointNetSetAbstraction_56478819942795
// MI455X (gfx1250) — compile-verified
//
#include <hip/hip_runtime.h>

// ---------------------------------------------------------------------------
// PointNet++ Set Abstraction for MI455X (gfx1250, wave32, WMMA f32 16x16x4)
// ---------------------------------------------------------------------------
// Pipeline (all on `stream`):
//   1) fps_kernel        : per-batch farthest point sampling -> new_xyz (d_out)
//   2) group_kernel      : ball query + gather + concat -> X0 (68 x 262144)
//   3) gemm_bn_kernel x3 : y = W*x + b via v_wmma_f32_16x16x4_f32 (templated,
//                          fully unrolled, LDS-staged weights, dual
//                          accumulators), previous-layer BN+ReLU fused on
//                          load, epilogue accumulates per-channel sum/sumsq
//   4) bn_finalize       : per-channel scale/shift from live batch statistics
//   5) maxpool_kernel    : feature[b,o,s] = max_k relu(y*scale+shift)
// ---------------------------------------------------------------------------

typedef float v2f __attribute__((ext_vector_type(2)));
typedef float v8f __attribute__((ext_vector_type(8)));

constexpr int    B_       = 16;
constexpr int    N_       = 2048;
constexpr int    S_       = 512;   // NPOINT
constexpr int    K_       = 32;    // NSAMPLE
constexpr size_t NCOLS    = (size_t)B_ * S_ * K_;   // 262144 GEMM columns
constexpr float  R2_      = 0.2f * 0.2f;

// -------------------------------- FPS --------------------------------------
__global__ void __launch_bounds__(256)
fps_kernel(const float* __restrict__ xyz, float* __restrict__ new_xyz)
{
    __shared__ float dist[N_];
    __shared__ float rv[256];
    __shared__ int   ri[256];
    __shared__ int   sfar;

    const int b = blockIdx.x;
    const int t = threadIdx.x;
    const float* xb = xyz + (size_t)b * 3 * N_;   // xb[c*N_ + n]

    for (int n = t; n < N_; n += 256) dist[n] = 1e10f;
    if (t == 0) sfar = 0;
    __syncthreads();

    for (int i = 0; i < S_; ++i) {
        const int far = sfar;
        const float cx = xb[far], cy = xb[N_ + far], cz = xb[2 * N_ + far];
        if (t == 0) {            // record selected point, transposed (B,3,S)
            new_xyz[(b * 3 + 0) * S_ + i] = cx;
            new_xyz[(b * 3 + 1) * S_ + i] = cy;
            new_xyz[(b * 3 + 2) * S_ + i] = cz;
        }
        float bv = -1.f; int bi = 0;
        for (int n = t; n < N_; n += 256) {
            const float dx = xb[n] - cx, dy = xb[N_ + n] - cy, dz = xb[2 * N_ + n] - cz;
            const float d  = dx * dx + dy * dy + dz * dz;
            const float dn = fminf(dist[n], d);
            dist[n] = dn;
            if (dn > bv) { bv = dn; bi = n; }   // strided n increasing -> first max kept
        }
        rv[t] = bv; ri[t] = bi;
        __syncthreads();
        for (int off = 128; off > 0; off >>= 1) {
            if (t < off) {
                const float v2 = rv[t + off]; const int i2 = ri[t + off];
                if (v2 > rv[t] || (v2 == rv[t] && i2 < ri[t])) { rv[t] = v2; ri[t] = i2; }
            }
            __syncthreads();
        }
        if (t == 0) sfar = ri[0];
        __syncthreads();
    }
}

// --------------------------- ball query + group -----------------------------
__global__ void __launch_bounds__(256)
group_kernel(const float* __restrict__ xyz, const float* __restrict__ pts,
             const float* __restrict__ new_xyz, float* __restrict__ X0)
{
    __shared__ int gidx[8][K_];
    const int lane = threadIdx.x & 31;
    const int w    = threadIdx.x >> 5;
    const int cent = blockIdx.x * 8 + w;            // 0 .. B_*S_-1
    const int b = cent >> 9, s = cent & (S_ - 1);

    const float* xb = xyz + (size_t)b * 3 * N_;
    const float cx = new_xyz[(b * 3 + 0) * S_ + s];
    const float cy = new_xyz[(b * 3 + 1) * S_ + s];
    const float cz = new_xyz[(b * 3 + 2) * S_ + s];

    int cnt = 0;
    for (int j0 = 0; j0 < N_ && cnt < K_; j0 += 32) {
        const int n = j0 + lane;
        const float dx = xb[n] - cx, dy = xb[N_ + n] - cy, dz = xb[2 * N_ + n] - cz;
        const bool inside = (dx * dx + dy * dy + dz * dz) <= R2_;
        const unsigned mask = (unsigned)__ballot(inside);
        const int p = cnt + __popc(mask & ((1u << lane) - 1u));
        if (inside && p < K_) gidx[w][p] = n;
        cnt += __popc(mask);
    }
    __syncthreads();

    const int total = cnt < K_ ? cnt : K_;          // >=1 (centroid is its own point)
    const int idx = (lane < total) ? gidx[w][lane] : gidx[w][0];

    const size_t col = (size_t)cent * K_ + lane;    // lane == neighbor k
    X0[0 * NCOLS + col] = xb[idx] - cx;             // grouped_xyz_norm
    X0[1 * NCOLS + col] = xb[N_ + idx] - cy;
    X0[2 * NCOLS + col] = xb[2 * N_ + idx] - cz;
    const float* pb = pts + (size_t)b * 64 * N_;
    #pragma unroll 4
    for (int c = 0; c < 64; ++c)
        X0[(size_t)(3 + c) * NCOLS + col] = pb[(size_t)c * N_ + idx];
    X0[(size_t)67 * NCOLS + col] = 0.f;             // K-pad row (67 -> 68)
}

// ------------------------------ GEMM + BN stats -----------------------------
// One 16(o) x 16(col) tile per wave via v_wmma_f32_16x16x4_f32; 8 waves/block.
// CIN   : true input-channel count (W row length)
// KSTEPS: ceil(CIN/4); X buffer must have KSTEPS*4 rows (extra rows zeroed)
// XFORM : apply relu(x*inscale+inshift) to X on load (previous layer's BN)
template <int CIN, int KSTEPS, bool XFORM>
__global__ void __launch_bounds__(256)
gemm_bn_kernel(const float* __restrict__ X, const float* __restrict__ W,
               const float* __restrict__ bias,
               const float* __restrict__ inscale, const float* __restrict__ inshift,
               float* __restrict__ Y, float* __restrict__ gsum, float* __restrict__ gsq)
{
    constexpr int CPAD = KSTEPS * 4;
    __shared__ float sw[16 * CPAD];       // zero-padded 16 x CPAD weight tile
    __shared__ float ssc[CPAD], ssh[CPAD];
    __shared__ float lsum[16], lsq[16];

    const int t = threadIdx.x;
    const int obase = blockIdx.y << 4;

    // cooperative load of the padded weight tile (and BN transform vectors)
    for (int e = t; e < 16 * CPAD; e += 256) {
        const int o = e / CPAD, k = e - o * CPAD;
        sw[e] = (k < CIN) ? W[(size_t)(obase + o) * CIN + k] : 0.f;
    }
    if (XFORM) {
        if (t < CPAD) { ssc[t] = inscale[t]; ssh[t] = inshift[t]; }
    }
    if (t < 16) { lsum[t] = 0.f; lsq[t] = 0.f; }
    __syncthreads();

    const int lane = t & 31, wave = t >> 5;
    const int half = lane >> 4, nn = lane & 15;
    const size_t col = (size_t)blockIdx.x * 128 + wave * 16 + nn;

    // preload A fragments: lane half 0 -> K={4k,4k+1}, half 1 -> K={4k+2,4k+3}
    v2f afrag[KSTEPS];
    {
        const v2f* swv = (const v2f*)sw;
        const int base2 = (nn * CPAD + half * 2) >> 1;   // in v2f units
        #pragma unroll
        for (int ks = 0; ks < KSTEPS; ++ks)
            afrag[ks] = swv[base2 + ks * 2];
    }

    v8f acc0, acc1;
    #pragma unroll
    for (int i = 0; i < 8; ++i) { acc0[i] = bias[obase + half * 8 + i]; acc1[i] = 0.f; }

    const v2f* scv = (const v2f*)ssc;
    const v2f* shv = (const v2f*)ssh;

    #pragma unroll
    for (int ks = 0; ks < KSTEPS; ++ks) {
        const int k = ks * 4 + half * 2;
        float b0 = X[(size_t)k * NCOLS + col];
        float b1 = X[(size_t)(k + 1) * NCOLS + col];
        if (XFORM) {                                   // fused previous BN + ReLU
            const v2f sc = scv[k >> 1];
            const v2f sh = shv[k >> 1];
            b0 = fmaxf(fmaf(b0, sc.x, sh.x), 0.f);
            b1 = fmaxf(fmaf(b1, sc.y, sh.y), 0.f);
        }
        v2f bf; bf.x = b0; bf.y = b1;
        if (ks & 1)
            acc1 = __builtin_amdgcn_wmma_f32_16x16x4_f32(
                       false, afrag[ks], false, bf, (short)0, acc1, false, false);
        else
            acc0 = __builtin_amdgcn_wmma_f32_16x16x4_f32(
                       false, afrag[ks], false, bf, (short)0, acc0, false, false);
    }
    acc0 = acc0 + acc1;

    // epilogue: store raw y, accumulate per-channel sum / sumsq for BatchNorm
    #pragma unroll
    for (int i = 0; i < 8; ++i) {
        const int o = obase + half * 8 + i;
        const float y = acc0[i];
        Y[(size_t)o * NCOLS + col] = y;
        float s1 = y, s2 = y * y;
        #pragma unroll
        for (int off = 1; off < 16; off <<= 1) {    // reduce across 16 columns of this half
            s1 += __shfl_xor(s1, off, 32);
            s2 += __shfl_xor(s2, off, 32);
        }
        if (nn == 0) {
            atomicAdd(&lsum[half * 8 + i], s1);
            atomicAdd(&lsq [half * 8 + i], s2);
        }
    }
    __syncthreads();
    if (t < 16) {
        atomicAdd(&gsum[obase + t], lsum[t]);
        atomicAdd(&gsq [obase + t], lsq [t]);
    }
}

// ------------------------------ BN helpers ----------------------------------
__global__ void zero_stats_kernel(float* gsum, float* gsq)
{
    const int t = threadIdx.x;
    if (t < 128) { gsum[t] = 0.f; gsq[t] = 0.f; }
}

__global__ void bn_finalize_kernel(const float* __restrict__ gsum, const float* __restrict__ gsq,
                                   const float* __restrict__ gamma, const float* __restrict__ beta,
                                   float* __restrict__ scale, float* __restrict__ shift, int O)
{
    const int o = threadIdx.x;
    if (o >= O) return;
    const float invN = 1.0f / (float)NCOLS;
    const float mean = gsum[o] * invN;
    const float var  = gsq[o] * invN - mean * mean;
    const float a    = gamma[o] * rsqrtf(var + 1e-5f);
    scale[o] = a;
    shift[o] = beta[o] - mean * a;
}

// --------------------------- BN + ReLU + max over k -------------------------
__global__ void __launch_bounds__(256)
maxpool_kernel(const float* __restrict__ Y3, const float* __restrict__ scale,
               const float* __restrict__ shift, float* __restrict__ feat)
{
    const int t = blockIdx.x * 256 + threadIdx.x;   // 16*128*512 threads
    const int s = t & (S_ - 1);
    const int o = (t >> 9) & 127;
    const int b = t >> 16;
    const float a = scale[o], c = shift[o];
    const float4* p = (const float4*)(Y3 + (size_t)o * NCOLS + (size_t)(b * S_ + s) * K_);
    // relu is monotone, a constant over k: max_k relu(a*y+c) = relu(a*ext_k y + c)
    // where ext = max if a>=0 else min.
    float mx = -3.4e38f, mn = 3.4e38f;
    #pragma unroll
    for (int j = 0; j < 8; ++j) {
        const float4 v = p[j];
        mx = fmaxf(mx, fmaxf(fmaxf(v.x, v.y), fmaxf(v.z, v.w)));
        mn = fminf(mn, fminf(fminf(v.x, v.y), fminf(v.z, v.w)));
    }
    const float raw = (a >= 0.f) ? fmaf(a, mx, c) : fmaf(a, mn, c);
    feat[(size_t)(b * 128 + o) * S_ + s] = fmaxf(raw, 0.f);
}

// ------------------------------- launcher -----------------------------------
extern "C" void kernel_launch(void* const* d_in, const int* in_sizes, int n_in,
                              void* d_out, int out_size, void* d_ws, size_t ws_size,
                              hipStream_t stream)
{
    (void)in_sizes; (void)n_in; (void)out_size; (void)ws_size;

    const float* xyz = (const float*)d_in[0];   // (16,3,2048)
    const float* pts = (const float*)d_in[1];   // (16,64,2048)
    const float* w0  = (const float*)d_in[2];   // (64,67)
    const float* b0  = (const float*)d_in[3];
    const float* g0  = (const float*)d_in[4];
    const float* bt0 = (const float*)d_in[5];
    const float* w1  = (const float*)d_in[6];   // (64,64)
    const float* b1  = (const float*)d_in[7];
    const float* g1  = (const float*)d_in[8];
    const float* bt1 = (const float*)d_in[9];
    const float* w2  = (const float*)d_in[10];  // (128,64)
    const float* b2  = (const float*)d_in[11];
    const float* g2  = (const float*)d_in[12];
    const float* bt2 = (const float*)d_in[13];

    float* out     = (float*)d_out;
    float* new_xyz = out;                       // (16,3,512)  = 24576 floats
    float* feat    = out + (size_t)B_ * 3 * S_; // (16,128,512)

    // workspace layout
    float* bufA = (float*)d_ws;                        // 68  x NCOLS (X0, then Y2)
    float* bufB = bufA + (size_t)68 * NCOLS;           // 64  x NCOLS (Y1)
    float* bufC = bufB + (size_t)64 * NCOLS;           // 128 x NCOLS (Y3)
    float* gsum = bufC + (size_t)128 * NCOLS;          // 128
    float* gsq  = gsum + 128;
    float* sc0  = gsq  + 128;  float* sh0 = sc0 + 128;
    float* sc1  = sh0  + 128;  float* sh1 = sc1 + 128;
    float* sc2  = sh1  + 128;  float* sh2 = sc2 + 128;

    // 1) FPS -> new_xyz (also first output)
    fps_kernel<<<B_, 256, 0, stream>>>(xyz, new_xyz);

    // 2) ball query + grouping -> X0 (68 rows, row 67 zero)
    group_kernel<<<(B_ * S_) / 8, 256, 0, stream>>>(xyz, pts, new_xyz, bufA);

    // 3) layer 0: 67 -> 64
    zero_stats_kernel<<<1, 128, 0, stream>>>(gsum, gsq);
    gemm_bn_kernel<67, 17, false><<<dim3(NCOLS / 128, 64 / 16), 256, 0, stream>>>(
        bufA, w0, b0, nullptr, nullptr, bufB, gsum, gsq);
    bn_finalize_kernel<<<1, 128, 0, stream>>>(gsum, gsq, g0, bt0, sc0, sh0, 64);

    // 4) layer 1: 64 -> 64 (BN0+ReLU fused into load)
    zero_stats_kernel<<<1, 128, 0, stream>>>(gsum, gsq);
    gemm_bn_kernel<64, 16, true><<<dim3(NCOLS / 128, 64 / 16), 256, 0, stream>>>(
        bufB, w1, b1, sc0, sh0, bufA, gsum, gsq);
    bn_finalize_kernel<<<1, 128, 0, stream>>>(gsum, gsq, g1, bt1, sc1, sh1, 64);

    // 5) layer 2: 64 -> 128 (BN1+ReLU fused into load)
    zero_stats_kernel<<<1, 128, 0, stream>>>(gsum, gsq);
    gemm_bn_kernel<64, 16, true><<<dim3(NCOLS / 128, 128 / 16), 256, 0, stream>>>(
        bufA, w2, b2, sc1, sh1, bufC, gsum, gsq);
    bn_finalize_kernel<<<1, 128, 0, stream>>>(gsum, gsq, g2, bt2, sc2, sh2, 128);

    // 6) BN2 + ReLU + max over nsample -> feature
    maxpool_kernel<<<(B_ * 128 * S_) / 256, 256, 0, stream>>>(bufC, sc2, sh2, feat);
}